// OperationLayer_40596030882205
// MI455X (gfx1250) — compile-verified
//
#include <hip/hip_runtime.h>
#include <hip/hip_bf16.h>

// ---- problem constants --------------------------------------------------
#define Bn 4
#define Cn 64
#define Hn 64
#define Wn 64
#define HWn 4096         // Hn*Wn
#define OPN 8
#define K1 512           // Cn*OPN
#define RK 8

typedef __attribute__((ext_vector_type(2))) float v2f;
typedef __attribute__((ext_vector_type(8))) float v8f;

struct Ptr7 { const float* p[7]; };

// ---- A_eff[b][m][op*64+c'] = w[b,op] * sum_c Wcp1flat[m][op*64+c] * PW_op[c][c']
// Wcp1flat[m=(r*8+k)][c512] = w_cp1[r][c512][k]  (w_cp1 is [8,512,8] row-major)
__global__ void k_prepA(const float* __restrict__ wcp1, const float* __restrict__ wts,
                        Ptr7 pws, float* __restrict__ Aeff) {
    int idx = blockIdx.x * blockDim.x + threadIdx.x;      // B*8*4096 = 131072
    if (idx >= Bn * OPN * 4096) return;
    int mc = idx & 4095;
    int op = (idx >> 12) & 7;
    int b  = idx >> 15;
    int m  = mc >> 6, cp = mc & 63;
    int r = m >> 3, kk = m & 7;
    const float* wc = wcp1 + r * (K1 * RK) + kk;          // stride 8 over c512
    float wgt = wts[b * OPN + op];
    float acc;
    if (op < 7) {
        const float* pw = pws.p[op];
        acc = 0.f;
        for (int c = 0; c < Cn; ++c)
            acc += wc[(op * Cn + c) * RK] * pw[c * Cn + cp];
    } else {
        acc = wc[(7 * Cn + cp) * RK];                     // pool op: no pointwise
    }
    Aeff[((long)b * 64 + m) * K1 + op * Cn + cp] = acc * wgt;
}

// ---- W3f[o][q=r*8+p] = w_out3[p][o][r]   (w_out3 is [8,64,8] row-major)
__global__ void k_prepW3(const float* __restrict__ wout3, float* __restrict__ W3f) {
    int idx = blockIdx.x * blockDim.x + threadIdx.x;      // 4096
    if (idx >= 4096) return;
    int o = idx >> 6, q = idx & 63;
    int r = q >> 3, p = q & 7;
    W3f[idx] = wout3[p * (Cn * RK) + o * RK + r];
}

// ---- depthwise convs on relu(x) + avg-pool -> S[b][op*64+c][pix] -------
__global__ void k_dw(const float* __restrict__ x, Ptr7 dws, float* __restrict__ S) {
    int idx = blockIdx.x * blockDim.x + threadIdx.x;      // B*C*HW = 1,048,576
    if (idx >= Bn * Cn * HWn) return;
    int pix = idx & (HWn - 1);
    int c   = (idx >> 12) & (Cn - 1);
    int b   = idx >> 18;
    int h = pix >> 6, w = pix & 63;
    const float* xp = x + ((long)b * Cn + c) * HWn;
    long srow = ((long)b * K1 + c) * HWn + pix;           // + op*Cn*HWn per op

    const int KS[7]  = {1, 3, 5, 7, 3, 5, 7};
    const int DIL[7] = {1, 1, 1, 1, 2, 2, 2};
    for (int op = 0; op < 7; ++op) {
        int ks = KS[op], dil = DIL[op], rad = ks >> 1;
        const float* kp = dws.p[op] + c * ks * ks;
        float acc = 0.f;
        for (int i = 0; i < ks; ++i) {
            int yy = h + (i - rad) * dil;
            if ((unsigned)yy >= (unsigned)Hn) continue;
            for (int j = 0; j < ks; ++j) {
                int xx = w + (j - rad) * dil;
                if ((unsigned)xx >= (unsigned)Wn) continue;
                float v = xp[yy * Wn + xx];
                acc += kp[i * ks + j] * (v > 0.f ? v : 0.f);
            }
        }
        S[srow + (long)op * Cn * HWn] = acc;
    }
    // avg_pool3 (count_include_pad=False), no relu
    float acc = 0.f; int cnt = 0;
    for (int dy = -1; dy <= 1; ++dy) {
        int yy = h + dy; if ((unsigned)yy >= (unsigned)Hn) continue;
        for (int dx = -1; dx <= 1; ++dx) {
            int xx = w + dx; if ((unsigned)xx >= (unsigned)Wn) continue;
            acc += xp[yy * Wn + xx]; ++cnt;
        }
    }
    S[srow + 7L * Cn * HWn] = acc / (float)cnt;
}

// ---- WMMA fp32 GEMM: Out[b][64][HW] = A[b][64][K] x S[b][K][HW] --------
// block = 256 threads = 8 wave32s; wave (mt,nt): mt = wv&3 (M 0..63), nt = wv>>2 (N tile of 16)
// block covers M=64 x N=32; grid.x = HW/32, grid.z = B.
__global__ void k_gemm_wmma(const float* __restrict__ A, long aBatch,
                            const float* __restrict__ S, long sBatch, int K,
                            float* __restrict__ Out, long oBatch) {
    int b    = blockIdx.z;
    int lane = threadIdx.x & 31;
    int wv   = threadIdx.x >> 5;
    int m0   = (wv & 3) * 16;
    int n0   = blockIdx.x * 32 + (wv >> 2) * 16;
    int half = lane >> 4;          // which half-wave
    int lh   = lane & 15;

    const float* Ab = A + (long)b * aBatch;
    const float* Sb = S + (long)b * sBatch;
    // A fragment: lane holds row M = m0+lh, K-pair (half*2, half*2+1)
    const float* arow = Ab + (long)(m0 + lh) * K + half * 2;
    // B fragment: lane holds column N = n0+lh, same K-pair rows
    const float* scol = Sb + n0 + lh;

    v8f acc = {};
#pragma unroll 8
    for (int k0 = 0; k0 < K; k0 += 4) {
        __builtin_prefetch(arow + k0 + 64, 0, 1);
        v2f a, bb;
        a.x  = arow[k0];
        a.y  = arow[k0 + 1];
        int kb = k0 + half * 2;
        bb.x = scol[(long)kb * HWn];
        bb.y = scol[(long)(kb + 1) * HWn];
        acc = __builtin_amdgcn_wmma_f32_16x16x4_f32(
            /*neg_a=*/false, a, /*neg_b=*/false, bb,
            /*c_mod=*/(short)0, acc, /*reuse_a=*/false, /*reuse_b=*/false);
    }
    // D layout: VGPR v -> M = m0 + v + half*8, N = n0 + lh
    float* ob = Out + (long)b * oBatch + n0 + lh;
#pragma unroll
    for (int v = 0; v < 8; ++v)
        ob[(long)(m0 + v + half * 8) * HWn] = acc[v];
}

// ---- per-pixel M^2: P[q=r*8+p] = sum_k out1[r*8+k] * out1[k*8+p] -------
__global__ void k_msq(const float* __restrict__ out1, float* __restrict__ P) {
    int idx = blockIdx.x * blockDim.x + threadIdx.x;      // B*HW = 16384
    if (idx >= Bn * HWn) return;
    int pix = idx & (HWn - 1);
    int b   = idx >> 12;
    const float* ip = out1 + (long)b * 64 * HWn + pix;
    float m[64];
#pragma unroll
    for (int i = 0; i < 64; ++i) m[i] = ip[(long)i * HWn];
    float* op = P + (long)b * 64 * HWn + pix;
#pragma unroll
    for (int r = 0; r < 8; ++r)
#pragma unroll
        for (int p = 0; p < 8; ++p) {
            float acc = 0.f;
#pragma unroll
            for (int k = 0; k < 8; ++k) acc += m[r * 8 + k] * m[k * 8 + p];
            op[(long)(r * 8 + p) * HWn] = acc;
        }
}

// ---- instance norm (biased var) + softsign; one block per (b,c) --------
__global__ void k_norm(const float* __restrict__ Y, float* __restrict__ outp) {
    int bc = blockIdx.x;                                  // 0..255
    const float* yp = Y + (long)bc * HWn;
    float* op = outp + (long)bc * HWn;
    __shared__ float s1[256], s2[256];
    float a1 = 0.f, a2 = 0.f;
    for (int i = threadIdx.x; i < HWn; i += 256) { float v = yp[i]; a1 += v; a2 += v * v; }
    s1[threadIdx.x] = a1; s2[threadIdx.x] = a2;
    __syncthreads();
    for (int s = 128; s > 0; s >>= 1) {
        if ((int)threadIdx.x < s) {
            s1[threadIdx.x] += s1[threadIdx.x + s];
            s2[threadIdx.x] += s2[threadIdx.x + s];
        }
        __syncthreads();
    }
    float mean = s1[0] * (1.f / HWn);
    float var  = s2[0] * (1.f / HWn) - mean * mean;
    float inv  = rsqrtf(var + 1e-5f);
    for (int i = threadIdx.x; i < HWn; i += 256) {
        float yn = (yp[i] - mean) * inv;
        op[i] = yn / (1.f + fabsf(yn));
    }
}

extern "C" void kernel_launch(void* const* d_in, const int* in_sizes, int n_in,
                              void* d_out, int out_size, void* d_ws, size_t ws_size,
                              hipStream_t stream) {
    const float* x     = (const float*)d_in[0];
    const float* wts   = (const float*)d_in[1];
    Ptr7 dws, pws;
    for (int i = 0; i < 7; ++i) dws.p[i] = (const float*)d_in[2 + i];   // dw1,dw3,dw5,dw7,ddw3,ddw5,ddw7
    for (int i = 0; i < 7; ++i) pws.p[i] = (const float*)d_in[9 + i];   // pw1,pw3,pw5,pw7,dpw3,dpw5,dpw7
    const float* wcp1  = (const float*)d_in[16];
    const float* wout3 = (const float*)d_in[17];
    float* outp = (float*)d_out;

    // workspace layout (floats)
    float* ws   = (float*)d_ws;
    float* S    = ws;                                  // B*512*HW = 8,388,608
    float* Aeff = S    + (long)Bn * K1 * HWn;          // B*64*512 = 131,072
    float* W3f  = Aeff + (long)Bn * 64 * K1;           // 4096
    float* out1 = W3f  + 4096;                         // B*64*HW = 1,048,576
    float* Pbuf = out1 + (long)Bn * 64 * HWn;          // B*64*HW
    float* ypre = Pbuf + (long)Bn * 64 * HWn;          // B*64*HW

    // 1. fold pointwise convs + arch weights into per-batch A_eff; pack w_out3
    k_prepA<<<(Bn * OPN * 4096 + 255) / 256, 256, 0, stream>>>(wcp1, wts, pws, Aeff);
    k_prepW3<<<16, 256, 0, stream>>>(wout3, W3f);

    // 2. depthwise convs (relu-fused) + avg pool -> S[b][512][HW]
    k_dw<<<(Bn * Cn * HWn + 255) / 256, 256, 0, stream>>>(x, dws, S);

    // 3. out1[b][64][HW] = A_eff[b] (64x512) x S[b] (512xHW)   [WMMA fp32]
    k_gemm_wmma<<<dim3(HWn / 32, 1, Bn), 256, 0, stream>>>(
        Aeff, (long)64 * K1, S, (long)K1 * HWn, K1, out1, (long)64 * HWn);

    // 4. per-pixel 8x8 M^2 -> P
    k_msq<<<(Bn * HWn + 255) / 256, 256, 0, stream>>>(out1, Pbuf);

    // 5. ypre[b][64][HW] = W3f (64x64) x P[b] (64xHW)          [WMMA fp32]
    k_gemm_wmma<<<dim3(HWn / 32, 1, Bn), 256, 0, stream>>>(
        W3f, 0L, Pbuf, (long)64 * HWn, 64, ypre, (long)64 * HWn);

    // 6. instance norm + softsign -> output [B,C,H,W]
    k_norm<<<Bn * Cn, 256, 0, stream>>>(ypre, outp);
}